// SinkhornKDLoss_73323681677451
// MI455X (gfx1250) — compile-verified
//
#include <hip/hip_runtime.h>
#include <math.h>

// Problem constants (match reference setup_inputs / reference()).
#define REG_F 0.1f
#define EPS_F 1e-8f
static constexpr int B_  = 8;
static constexpr int LS_ = 1024;
static constexpr int LT_ = 1024;
static constexpr int D_  = 768;

typedef __attribute__((ext_vector_type(2))) float v2f;
typedef __attribute__((ext_vector_type(8))) float v8f;

// ---------------------------------------------------------------------------
// 1) Squared norms: one wave (32 lanes) per token, shuffle reduction.
// ---------------------------------------------------------------------------
__global__ void sqnorm_kernel(const float* __restrict__ x,
                              float* __restrict__ out, int ntok) {
    int wave = (blockIdx.x * blockDim.x + threadIdx.x) >> 5;
    int lane = threadIdx.x & 31;
    if (wave >= ntok) return;
    const float* row = x + (size_t)wave * D_;
    float acc = 0.0f;
    for (int d = lane; d < D_; d += 32) {
        float v = row[d];
        acc = fmaf(v, v, acc);
    }
    #pragma unroll
    for (int off = 16; off > 0; off >>= 1)
        acc += __shfl_down(acc, off, 32);
    if (lane == 0) out[wave] = acc;
}

// ---------------------------------------------------------------------------
// 2) cdist + K via fp32 WMMA (V_WMMA_F32_16X16X4_F32).
//    Grid: (LS/16, LT/(16*8), B). Block: 256 threads = 8 waves.
//    Wave w of a workgroup computes tile (stile, blockIdx.y*8 + w): waves in a
//    WG share the same student tile (A) -> L0/L2 reuse. EXEC is all-ones
//    (no divergence) as WMMA requires.
//
//    f32 A (16x4, MxK) per ISA 7.12.2: lanes 0-15 -> M=lane, VGPR0=K0,VGPR1=K1;
//    lanes 16-31 -> M=lane-16, VGPR0=K2,VGPR1=K3.
//    f32 B (4x16, KxN): row striped across lanes in one VGPR, mirrored halves:
//    VGPR0: lanes0-15=K0 row (N=lane), lanes16-31=K2; VGPR1: K1/K3.
// ---------------------------------------------------------------------------
__global__ void cdist_wmma_kernel(const float* __restrict__ S,
                                  const float* __restrict__ T,
                                  const float* __restrict__ sq_s,
                                  const float* __restrict__ sq_t,
                                  float* __restrict__ Cmat,
                                  float* __restrict__ Kmat) {
    const int b     = blockIdx.z;
    const int stile = blockIdx.x;                    // 0..63
    const int wave  = threadIdx.x >> 5;              // 0..7
    const int ttile = blockIdx.y * 8 + wave;         // 0..63
    const int lane  = threadIdx.x & 31;

    const float* Sb = S + (size_t)b * LS_ * D_;
    const float* Tb = T + (size_t)b * LT_ * D_;

    const int mn    = lane & 15;                     // M for A, N for B
    const int khalf = (lane >> 4) << 1;              // 0 or 2

    // 8-byte aligned: row*768 even, khalf even, d multiple of 4.
    const float* arow = Sb + (size_t)(stile * 16 + mn) * D_ + khalf;
    const float* brow = Tb + (size_t)(ttile * 16 + mn) * D_ + khalf;

    v8f acc = {};
    for (int d = 0; d < D_; d += 4) {
        v2f a  = *(const v2f*)(arow + d);
        v2f bb = *(const v2f*)(brow + d);
        acc = __builtin_amdgcn_wmma_f32_16x16x4_f32(
            /*neg_a=*/false, a, /*neg_b=*/false, bb,
            /*c_mod=*/(short)0, acc, /*reuse_a=*/false, /*reuse_b=*/false);
    }

    // Epilogue: C/D layout -> VGPR i holds M = i (+8 for upper half), N = lane&15.
    const int   gt    = ttile * 16 + mn;
    const float sqt   = sq_t[(size_t)b * LT_ + gt];
    const int   mbase = stile * 16 + ((lane >> 4) << 3);
    float* Cb = Cmat + (size_t)b * LS_ * LT_;
    float* Kb = Kmat + (size_t)b * LS_ * LT_;
    #pragma unroll
    for (int i = 0; i < 8; ++i) {
        int   gs = mbase + i;
        float d2 = sq_s[(size_t)b * LS_ + gs] + sqt - 2.0f * acc[i];
        float Cv = sqrtf(fmaxf(d2, 0.0f));
        float Kv = expf(-Cv * (1.0f / REG_F));
        size_t off = (size_t)gs * LT_ + gt;
        Cb[off] = Cv;
        Kb[off] = Kv;
    }
}

// ---------------------------------------------------------------------------
// 3a) init u = 1
// ---------------------------------------------------------------------------
__global__ void ones_kernel(float* __restrict__ p, int n) {
    int i = blockIdx.x * blockDim.x + threadIdx.x;
    if (i < n) p[i] = 1.0f;
}

// ---------------------------------------------------------------------------
// 3b) v[b,t] = (1/LT) / (sum_s K[b,s,t]*u[b,s] + eps)
//     Thread-per-column; coalesced K reads; u staged in LDS.
//     Grid: (LT/256, B), block 256.
// ---------------------------------------------------------------------------
__global__ void v_update_kernel(const float* __restrict__ Kmat,
                                const float* __restrict__ u,
                                float* __restrict__ v) {
    __shared__ float su[LS_];
    const int b = blockIdx.y;
    const int t = blockIdx.x * blockDim.x + threadIdx.x;
    const float* ub = u + (size_t)b * LS_;
    for (int i = threadIdx.x; i < LS_; i += blockDim.x) su[i] = ub[i];
    __syncthreads();
    const float* Kb = Kmat + (size_t)b * LS_ * LT_;
    float acc = 0.0f;
    for (int s = 0; s < LS_; ++s)
        acc = fmaf(Kb[(size_t)s * LT_ + t], su[s], acc);
    v[(size_t)b * LT_ + t] = (1.0f / (float)LT_) / (acc + EPS_F);
}

// ---------------------------------------------------------------------------
// 3c) u[b,s] = (1/LS) / (sum_t K[b,s,t]*v[b,t] + eps)
//     Workgroup-per-row, deterministic tree reduction. Grid: B*LS blocks.
// ---------------------------------------------------------------------------
__global__ void u_update_kernel(const float* __restrict__ Kmat,
                                const float* __restrict__ v,
                                float* __restrict__ u) {
    __shared__ float red[256];
    const int row = blockIdx.x;            // 0..B*LS-1
    const int b   = row >> 10;
    const float* Kr = Kmat + (size_t)row * LT_;
    const float* vb = v + (size_t)b * LT_;
    float acc = 0.0f;
    for (int t = threadIdx.x; t < LT_; t += blockDim.x)
        acc = fmaf(Kr[t], vb[t], acc);
    red[threadIdx.x] = acc;
    __syncthreads();
    #pragma unroll
    for (int s = 128; s > 0; s >>= 1) {
        if ((int)threadIdx.x < s) red[threadIdx.x] += red[threadIdx.x + s];
        __syncthreads();
    }
    if (threadIdx.x == 0)
        u[row] = (1.0f / (float)LS_) / (red[0] + EPS_F);
}

// ---------------------------------------------------------------------------
// 4a) rowpart[row] = u[row] * sum_t K[row,t]*C[row,t]*v[b,t]
// ---------------------------------------------------------------------------
__global__ void loss_rows_kernel(const float* __restrict__ Cmat,
                                 const float* __restrict__ Kmat,
                                 const float* __restrict__ u,
                                 const float* __restrict__ v,
                                 float* __restrict__ rowpart) {
    __shared__ float red[256];
    const int row = blockIdx.x;
    const int b   = row >> 10;
    const float* Cr = Cmat + (size_t)row * LT_;
    const float* Kr = Kmat + (size_t)row * LT_;
    const float* vb = v + (size_t)b * LT_;
    float acc = 0.0f;
    for (int t = threadIdx.x; t < LT_; t += blockDim.x)
        acc = fmaf(Kr[t] * Cr[t], vb[t], acc);
    red[threadIdx.x] = acc;
    __syncthreads();
    #pragma unroll
    for (int s = 128; s > 0; s >>= 1) {
        if ((int)threadIdx.x < s) red[threadIdx.x] += red[threadIdx.x + s];
        __syncthreads();
    }
    if (threadIdx.x == 0)
        rowpart[row] = u[row] * red[0];
}

// ---------------------------------------------------------------------------
// 4b) single-block deterministic sum of B*LS row partials, /B -> d_out[0]
// ---------------------------------------------------------------------------
__global__ void loss_final_kernel(const float* __restrict__ rowpart,
                                  float* __restrict__ out) {
    __shared__ float red[256];
    float acc = 0.0f;
    for (int i = threadIdx.x; i < B_ * LS_; i += 256)
        acc += rowpart[i];
    red[threadIdx.x] = acc;
    __syncthreads();
    #pragma unroll
    for (int s = 128; s > 0; s >>= 1) {
        if ((int)threadIdx.x < s) red[threadIdx.x] += red[threadIdx.x + s];
        __syncthreads();
    }
    if (threadIdx.x == 0)
        out[0] = red[0] / (float)B_;
}

// ---------------------------------------------------------------------------
// Launcher
// ---------------------------------------------------------------------------
extern "C" void kernel_launch(void* const* d_in, const int* in_sizes, int n_in,
                              void* d_out, int out_size, void* d_ws, size_t ws_size,
                              hipStream_t stream) {
    const float* S = (const float*)d_in[0];   // (B, LS, D) fp32
    const float* T = (const float*)d_in[1];   // (B, LT, D) fp32
    float* out = (float*)d_out;

    // Workspace layout (floats): C | K | sq_s | sq_t | u | v | rowpart
    const size_t matN = (size_t)B_ * LS_ * LT_;     // 8M floats each
    float* ws      = (float*)d_ws;
    float* Cmat    = ws;
    float* Kmat    = Cmat + matN;
    float* sq_s    = Kmat + matN;
    float* sq_t    = sq_s + (size_t)B_ * LS_;
    float* u       = sq_t + (size_t)B_ * LT_;
    float* v       = u    + (size_t)B_ * LS_;
    float* rowpart = v    + (size_t)B_ * LT_;

    // 1) squared norms (8 waves/block -> 8 tokens/block)
    sqnorm_kernel<<<dim3((B_ * LS_) / 8), 256, 0, stream>>>(S, sq_s, B_ * LS_);
    sqnorm_kernel<<<dim3((B_ * LT_) / 8), 256, 0, stream>>>(T, sq_t, B_ * LT_);

    // 2) C and K via fp32 WMMA
    cdist_wmma_kernel<<<dim3(LS_ / 16, LT_ / (16 * 8), B_), 256, 0, stream>>>(
        S, T, sq_s, sq_t, Cmat, Kmat);

    // 3) Sinkhorn: u = 1; 10x { v-update; u-update }
    ones_kernel<<<dim3((B_ * LS_ + 255) / 256), 256, 0, stream>>>(u, B_ * LS_);
    for (int it = 0; it < 10; ++it) {
        v_update_kernel<<<dim3(LT_ / 256, B_), 256, 0, stream>>>(Kmat, u, v);
        u_update_kernel<<<dim3(B_ * LS_), 256, 0, stream>>>(Kmat, v, u);
    }

    // 4) loss = mean_b sum_{s,t} u*K*C*v  (two-stage deterministic reduction)
    loss_rows_kernel<<<dim3(B_ * LS_), 256, 0, stream>>>(Cmat, Kmat, u, v, rowpart);
    loss_final_kernel<<<dim3(1), 256, 0, stream>>>(rowpart, out);

    (void)in_sizes; (void)n_in; (void)out_size; (void)ws_size;
}